// DNN_WPE_7765300871172
// MI455X (gfx1250) — compile-verified
//
#include <hip/hip_runtime.h>

typedef __attribute__((ext_vector_type(2))) float v2f;
typedef __attribute__((ext_vector_type(8))) float v8f;

#define WPE_TAPS 5
#define WPE_DELAY 3

namespace {
constexpr int NB = 8, NT = 600, NC = 8, NF = 257;
constexpr int TQ = NT - WPE_DELAY - WPE_TAPS + 1; // 593 valid frames
constexpr int KCHUNKS = (TQ + 3) / 4;             // 149 K-steps of 4
constexpr int TPAD = 608;                          // padded time axis in LDS
constexpr int NROW = WPE_TAPS * NC;                // 40
constexpr int NAUG = NROW + NC;                    // 48 (R | P augmented)
constexpr int NTHREADS = 288;                      // 9 waves -> 3x3 WMMA tiles
constexpr float WPE_EPS = 1e-10f;
}

#define WMMA_F32(A, B, C)                                                     \
  __builtin_amdgcn_wmma_f32_16x16x4_f32(false, (A), false, (B), (short)0,     \
                                        (C), false, false)

// One workgroup per (b, f) problem. Everything staged in LDS (~99 KB < 320 KB).
__global__ __launch_bounds__(NTHREADS) void wpe_kernel(
    const float* __restrict__ dre, const float* __restrict__ dimg,
    const int* __restrict__ ilens, float* __restrict__ outRe,
    float* __restrict__ outIm) {
  __shared__ float Yre[NC][TPAD], Yim[NC][TPAD];   // original observation
  __shared__ float Ere[NC][TPAD], Eim[NC][TPAD];   // enhanced estimate
  __shared__ float invp[TPAD];
  __shared__ float Mre[NAUG][NAUG + 1], Mim[NAUG][NAUG + 1]; // [R | P] system
  __shared__ float colRe[NROW], colIm[NROW];
  __shared__ float pivRe, pivIm;

  const int bf = blockIdx.x;
  const int b = bf / NF, f = bf % NF;
  const int tid = threadIdx.x;
  const int lane = tid & 31;
  const int wave = tid >> 5;

  // ---- stage Y(b, :, :, f) into LDS (strided gather; whole input lives in L2)
  for (int idx = tid; idx < NC * TPAD; idx += NTHREADS) {
    const int c = idx / TPAD, t = idx % TPAD;
    if (t < NT) {
      const size_t g = ((size_t)(b * NT + t) * NC + c) * NF + f;
      Yre[c][t] = dre[g];
      Yim[c][t] = dimg[g];
    } else {
      Yre[c][t] = 0.f;
      Yim[c][t] = 0.f;
    }
  }
  for (int t = NT + tid; t < TPAD; t += NTHREADS) invp[t] = 0.f;
  __syncthreads();

  // ---- WMMA tile assignment: wave -> 16x16 tile of the 48x48 output
  const int mt = wave / 3, nt = wave % 3;
  const int mloc = lane & 15;    // M (A) / N (B) index within tile
  const int hi = lane >> 4;      // K half selector per ISA 16x16x4 layout
  const int arow = mt * 16 + mloc;            // row r = k*C + d of A^H
  const int ak = (arow < NROW) ? (arow >> 3) : 0;  // rows 40..47: garbage, unread
  const int ad = arow & 7;
  const int bcol = nt * 16 + mloc;            // col: l*C+e (R) or 40+e (P)
  const bool bP = bcol >= NROW;
  const int bl = bP ? 0 : (bcol >> 3);
  const int be = bP ? (bcol - NROW) : (bcol & 7);

  // loop-invariant per-lane operand base pointers (hoisted out of K-loop)
  const float* const pAre = &Yre[ad][(WPE_TAPS - 1) - ak];
  const float* const pAim = &Yim[ad][(WPE_TAPS - 1) - ak];
  const int boff = bP ? (WPE_DELAY + WPE_TAPS - 1) : ((WPE_TAPS - 1) - bl);
  const float* const pBre = &Yre[be][boff];
  const float* const pBim = &Yim[be][boff];
  const float* const pW = &invp[WPE_DELAY + WPE_TAPS - 1];

  for (int it = 0; it < 2; ++it) {
    // ---- power over channels of current estimate; inv_p = 1/clip(power, eps)
    {
      float (*Sre)[TPAD] = (it == 0) ? Yre : Ere;
      float (*Sim)[TPAD] = (it == 0) ? Yim : Eim;
      for (int t = tid; t < NT; t += NTHREADS) {
        float s = 0.f;
        #pragma unroll
        for (int c = 0; c < NC; ++c)
          s += Sre[c][t] * Sre[c][t] + Sim[c][t] * Sim[c][t];
        s *= (1.0f / NC);
        invp[t] = 1.0f / fmaxf(s, WPE_EPS);
      }
    }
    __syncthreads();

    // ---- [R | P] = A^H * [B | Ynow] : complex GEMM via V_WMMA_F32_16X16X4_F32
    // A[r=(k*C+d)][t] = Y[d][t+4-k] * inv_p[t+7]  (conjugated -> negate Aim)
    // B[t][c=(l*C+e)] = Y[e][t+4-l] ; B[t][40+e] = Y[e][t+7]
    {
      v8f cre = {0.f, 0.f, 0.f, 0.f, 0.f, 0.f, 0.f, 0.f};
      v8f cim = {0.f, 0.f, 0.f, 0.f, 0.f, 0.f, 0.f, 0.f};
      // main body: chunks 0..147 are fully inside t < TQ -> no guards at all
      for (int k0 = 0; k0 < KCHUNKS - 1; ++k0) {
        const int t0 = 4 * k0 + 2 * hi;
        v2f ar, ai, br, bi;
        #pragma unroll
        for (int j = 0; j < 2; ++j) {
          const int t = t0 + j;
          const float w = pW[t];
          ar[j] = pAre[t] * w;
          ai[j] = pAim[t] * w;
          br[j] = pBre[t];
          bi[j] = pBim[t];
        }
        const v2f nai = -ai; // conj(A): f32 WMMA NEG is C-only, negate in VALU
        cre = WMMA_F32(ar, br, cre);
        cre = WMMA_F32(ai, bi, cre);
        cim = WMMA_F32(ar, bi, cim);
        cim = WMMA_F32(nai, br, cim);
      }
      // tail chunk 148: t in 592..595, only t < 593 contributes
      {
        const int t0 = 4 * (KCHUNKS - 1) + 2 * hi;
        v2f ar, ai, br, bi;
        #pragma unroll
        for (int j = 0; j < 2; ++j) {
          const int t = t0 + j;
          const bool tv = t < TQ;
          const float w = tv ? pW[t] : 0.f;  // LDS reads stay in padded bounds
          ar[j] = pAre[t] * w;
          ai[j] = pAim[t] * w;
          const float yrb = pBre[t], yib = pBim[t];
          br[j] = tv ? yrb : 0.f;
          bi[j] = tv ? yib : 0.f;
        }
        const v2f nai = -ai;
        cre = WMMA_F32(ar, br, cre);
        cre = WMMA_F32(ai, bi, cre);
        cim = WMMA_F32(ar, bi, cim);
        cim = WMMA_F32(nai, br, cim);
      }
      #pragma unroll
      for (int i = 0; i < 8; ++i) {  // C/D layout: VGPR i = row i / i+8
        const int row = mt * 16 + i + 8 * hi;
        const int col = nt * 16 + mloc;
        Mre[row][col] = cre[i];
        Mim[row][col] = cim[i];
      }
    }
    __syncthreads();
    if (tid < NROW) Mre[tid][tid] += WPE_EPS;  // R + eps*I
    __syncthreads();

    // ---- Gauss-Jordan on Hermitian-PD R: solve R X = P (cols 40..47)
    for (int j = 0; j < NROW; ++j) {
      if (tid == 0) {
        const float pr = Mre[j][j], pi = Mim[j][j];
        const float inv = 1.0f / (pr * pr + pi * pi);
        pivRe = pr * inv;
        pivIm = -pi * inv;
      }
      if (tid < NROW) {
        colRe[tid] = Mre[tid][j];
        colIm[tid] = Mim[tid][j];
      }
      __syncthreads();
      if (tid < NAUG) {  // scale pivot row
        const float xr = Mre[j][tid], xi = Mim[j][tid];
        Mre[j][tid] = xr * pivRe - xi * pivIm;
        Mim[j][tid] = xr * pivIm + xi * pivRe;
      }
      __syncthreads();
      for (int idx = tid; idx < NROW * NAUG; idx += NTHREADS) {
        const int i = idx / NAUG, cc = idx % NAUG;
        if (i == j) continue;
        const float fr = colRe[i], fi = colIm[i];
        const float rr = Mre[j][cc], ri = Mim[j][cc];
        Mre[i][cc] -= fr * rr - fi * ri;
        Mim[i][cc] -= fr * ri + fi * rr;
      }
      __syncthreads();
    }
    // X[l][e] now in Mre/Mim[l][40+e]; G_conj[p,d,e] = X[p*C+d][e]

    // ---- enhanced = Y - sum_{p,d} G_conj[p,d,e] * Y[d][t - DELAY - p]
    for (int idx = tid; idx < NC * NT; idx += NTHREADS) {
      const int e = idx / NT, t = idx % NT;
      float tr = 0.f, ti = 0.f;
      #pragma unroll
      for (int p = 0; p < WPE_TAPS; ++p) {
        const int ts = t - WPE_DELAY - p;
        if (ts < 0) continue;
        #pragma unroll
        for (int d = 0; d < NC; ++d) {
          const float gr = Mre[p * NC + d][NROW + e];
          const float gi = Mim[p * NC + d][NROW + e];
          const float yr = Yre[d][ts], yi = Yim[d][ts];
          tr += gr * yr - gi * yi;
          ti += gr * yi + gi * yr;
        }
      }
      Ere[e][t] = Yre[e][t] - tr;
      Eim[e][t] = Yim[e][t] - ti;
    }
    __syncthreads();
  }

  // ---- mask frames t >= ilens[b], scatter back to (B, T, C, F)
  const int len = ilens[b];
  for (int idx = tid; idx < NC * NT; idx += NTHREADS) {
    const int c = idx / NT, t = idx % NT;
    const size_t g = ((size_t)(b * NT + t) * NC + c) * NF + f;
    const float m = (t < len) ? 1.f : 0.f;
    outRe[g] = Ere[c][t] * m;
    outIm[g] = Eim[c][t] * m;
  }
}

extern "C" void kernel_launch(void* const* d_in, const int* in_sizes, int n_in,
                              void* d_out, int out_size, void* d_ws,
                              size_t ws_size, hipStream_t stream) {
  (void)in_sizes; (void)n_in; (void)d_ws; (void)ws_size; (void)out_size;
  const float* dre = (const float*)d_in[0];
  const float* dimg = (const float*)d_in[1];
  const int* ilens = (const int*)d_in[2];
  float* out = (float*)d_out;
  float* outIm = out + (size_t)NB * NT * NC * NF;
  wpe_kernel<<<dim3(NB * NF), dim3(NTHREADS), 0, stream>>>(dre, dimg, ilens,
                                                           out, outIm);
}